// Loss_40510131536268
// MI455X (gfx1250) — compile-verified
//
#include <hip/hip_runtime.h>
#include <math.h>

// ---------------------------------------------------------------------------
// Lattice forward loss for MI455X (gfx1250, wave32, WGP / 320 KB LDS).
//
// One persistent workgroup per batch element (B=32 blocks, 1024 thr = 32
// wave32). The T=256-step recurrence runs in-kernel with 3 barriers/step and
// zero global synchronization. Per-thread state: 16 packed edges + out-mask
// bits live in VGPRs for the whole kernel; LDS holds only the V=2048 state
// vectors (~27 KB).
//
// CDNA5 paths: global_load_async_to_lds_b64 (+ s_wait_asynccnt) single-buffers
// the per-step x row, global_prefetch_b8 covers t+1, ds_add_f32 (workgroup-
// scope relaxed fadd, non-returning) does the edge scatter-add.
// ---------------------------------------------------------------------------

static constexpr int   kV  = 2048;    // states
static constexpr int   kT  = 256;     // time steps
static constexpr int   kE  = 16384;   // edges
static constexpr int   kS  = 128;     // start/end nodes
static constexpr int   kNT = 1024;    // threads per block (32 wave32)
static constexpr int   kNW = kNT / 32;
static constexpr int   kEPT = kE / kNT;  // 16 edges per thread (in VGPRs)
static constexpr float kLogEps = -64.0f;
static constexpr float kEps    = 1.6038108905486378e-28f;  // exp(-64)

// exp that maps -inf and NaN (from -inf - -inf) to 0; args are always <= 0.
__device__ __forceinline__ float pexp(float z) {
  return (z > -87.0f) ? __expf(z) : 0.0f;
}

// Online logsumexp pair combine across a wave32 (1 exp per level):
// after this, every lane holds (m, s) with logsumexp = m + log(s).
__device__ __forceinline__ void wave_lse32(float& m, float& s) {
  #pragma unroll
  for (int off = 16; off > 0; off >>= 1) {
    const float m2 = __shfl_xor(m, off, 32);
    const float s2 = __shfl_xor(s, off, 32);
    const float d  = m - m2;              // NaN if both -inf -> e=0, pick s2(=0)
    const float e  = pexp(-fabsf(d));
    s = (d >= 0.0f) ? fmaf(s2, e, s) : fmaf(s, e, s2);
    m = fmaxf(m, m2);
  }
}
__device__ __forceinline__ float wave_max32(float v) {
  #pragma unroll
  for (int off = 16; off > 0; off >>= 1) v = fmaxf(v, __shfl_xor(v, off, 32));
  return v;
}
__device__ __forceinline__ float wave_sum32(float v) {
  #pragma unroll
  for (int off = 16; off > 0; off >>= 1) v += __shfl_xor(v, off, 32);
  return v;
}

// ---- layout transform: [V,B,T] -> [B,T,V] so each step reads 8 KB rows ----
__global__ __launch_bounds__(256) void transpose_vbt_to_btv(
    const float* __restrict__ in, float* __restrict__ out, int B) {
  __shared__ float tile[32][33];
  const int bv = blockIdx.x, bt = blockIdx.y, b = blockIdx.z;
  const int tx = threadIdx.x, ty = threadIdx.y;
  #pragma unroll
  for (int j = ty; j < 32; j += 8) {
    const int v = bv * 32 + j, t = bt * 32 + tx;
    tile[j][tx] = in[((size_t)v * B + b) * kT + t];      // coalesced in t
  }
  __syncthreads();
  #pragma unroll
  for (int j = ty; j < 32; j += 8) {
    const int t = bt * 32 + j, v = bv * 32 + tx;
    out[((size_t)b * kT + t) * kV + v] = tile[tx][j];    // coalesced in v
  }
}

// ---- main recurrence kernel ----------------------------------------------
__global__ __launch_bounds__(kNT) void lattice_forward(
    const float* __restrict__ x_raw,   // [V,B,T]
    const float* __restrict__ x_btv,   // [B,T,V] in workspace (if use_ws)
    const int*   __restrict__ tlen,    // [B]
    const int*   __restrict__ in_idx,  // [E] destination
    const int*   __restrict__ out_idx, // [E] source
    const int*   __restrict__ sidx,    // [S]
    const int*   __restrict__ eidx,    // [S]
    float*       __restrict__ outp,    // [B]
    int B, int use_ws) {
  __shared__ float          s_lp[kV];       //  8 KB  log_prev
  __shared__ float          s_comb[kV];     //  8 KB  scatter-add target
  __shared__ float          s_x[kV];        //  8 KB  async x-row buffer
  __shared__ unsigned char  s_mask[kV];     //  2 KB  out-node mask (setup only)
  __shared__ short          s_end[kS];
  __shared__ float          s_redm[kNW];
  __shared__ float          s_reds[kNW];

  const int tid  = threadIdx.x;
  const int lane = tid & 31;
  const int wid  = tid >> 5;
  const int b    = blockIdx.x;
  const int v0   = 2 * tid, v1 = 2 * tid + 1;
  const int L    = tlen[b];

  // ---- one-time setup: edges into VGPRs, out-mask built then hoisted ----
  s_mask[v0] = 0; s_mask[v1] = 0;
  unsigned edges[kEPT];                       // (dst<<16)|src, VGPR-resident
  #pragma unroll
  for (int k = 0; k < kEPT; ++k) {
    const int e = tid + k * kNT;
    edges[k] = (unsigned)out_idx[e] | ((unsigned)in_idx[e] << 16);
  }
  if (tid < kS) s_end[tid] = (short)eidx[tid];
  __syncthreads();
  #pragma unroll
  for (int k = 0; k < kEPT; ++k) s_mask[edges[k] & 0xffffu] = 1;

  // ---- t = 0 init ----
  if (use_ws) {
    const float* row = x_btv + (size_t)b * kT * kV;
    s_x[v0] = row[v0]; s_x[v1] = row[v1];
  } else {
    s_x[v0] = x_raw[((size_t)v0 * B + b) * kT];
    s_x[v1] = x_raw[((size_t)v1 * B + b) * kT];
  }
  s_lp[v0] = kLogEps; s_lp[v1] = kLogEps;
  __syncthreads();
  const bool mask0 = (s_mask[v0] != 0);       // hoist mask to registers
  const bool mask1 = (s_mask[v1] != 0);
  if (tid < kS) { const int si = sidx[tid]; s_lp[si] = s_x[si]; }
  __syncthreads();

  float acc = 0.0f, result = 0.0f;
  if (tid < 32) {                       // end-logsumexp at t=0, wave 0 only
    float vals[4];
    #pragma unroll
    for (int k = 0; k < 4; ++k) vals[k] = s_lp[s_end[tid + 32 * k]];
    float m = fmaxf(fmaxf(vals[0], vals[1]), fmaxf(vals[2], vals[3]));
    m = wave_max32(m);
    float s = 0.0f;
    #pragma unroll
    for (int k = 0; k < 4; ++k) s += pexp(vals[k] - m);
    s = wave_sum32(s);
    if (L == 1 && tid == 0) result = m + __logf(s) + acc;
  }

  // ---- sequential recurrence over t = 1..T-1 (3 barriers/step) ----
  for (int t = 1; t < kT; ++t) {
    // Phase A: launch this step's x-row fetch (consumed at phase D).
    float xr0 = 0.0f, xr1 = 0.0f;
    if (use_ws) {
      const float* row = x_btv + ((size_t)b * kT + t) * kV;
      const unsigned ldsa = (unsigned)(size_t)(&s_x[v0]);   // LDS byte offset
      const unsigned voff = (unsigned)(v0 * 4);             // 8B-aligned
      const unsigned long long base = (unsigned long long)(size_t)row;
      asm volatile("global_load_async_to_lds_b64 %0, %1, %2"
                   :: "v"(ldsa), "v"(voff), "s"(base) : "memory");
      if (t + 1 < kT && tid < (kV * 4) / 128) {             // cover t+1 row
        __builtin_prefetch(x_btv + ((size_t)b * kT + t + 1) * kV + tid * 32, 0, 0);
      }
    } else {
      xr0 = x_raw[((size_t)v0 * B + b) * kT + t];
      xr1 = x_raw[((size_t)v1 * B + b) * kT + t];
    }

    // Phase B: single-pass masked logsumexp over V -> log_C (one barrier).
    s_comb[v0] = 0.0f; s_comb[v1] = 0.0f;
    const float a0 = mask0 ? s_lp[v0] : -__builtin_inff();
    const float a1 = mask1 ? s_lp[v1] : -__builtin_inff();
    float m = fmaxf(a0, a1);
    float s = pexp(a0 - m) + pexp(a1 - m);
    wave_lse32(m, s);
    if (lane == 0) { s_redm[wid] = m; s_reds[wid] = s; }
    __syncthreads();                                        // barrier 1
    m = s_redm[lane]; s = s_reds[lane];
    wave_lse32(m, s);                    // all threads now hold (m, s)
    const float logC = m + __logf(s);
    acc += logC;

    // Phase C: edge propagation via ds_add_f32 (16 VGPR-resident edges/thread).
    #pragma unroll
    for (int k = 0; k < kEPT; ++k) {
      const unsigned pe = edges[k];
      const float z = s_lp[pe & 0xffffu] - logC;
      if (z > kLogEps) {                                    // exp_safe
        (void)__hip_atomic_fetch_add(&s_comb[pe >> 16], __expf(z),
                                     __ATOMIC_RELAXED, __HIP_MEMORY_SCOPE_WORKGROUP);
      }
    }

    // Phase D: fence async x-row, then combine.
    if (use_ws) {
#if __has_builtin(__builtin_amdgcn_s_wait_asynccnt)
      __builtin_amdgcn_s_wait_asynccnt(0);
#else
      asm volatile("s_wait_asynccnt 0" ::: "memory");
#endif
    }
    __syncthreads();                                        // barrier 2
    if (use_ws) { xr0 = s_x[v0]; xr1 = s_x[v1]; }
    const float c0 = s_comb[v0], c1 = s_comb[v1];
    const float l0 = ((c0 > kEps) ? __logf(c0) : kLogEps) + xr0;  // log_safe
    const float l1 = ((c1 > kEps) ? __logf(c1) : kLogEps) + xr1;
    s_lp[v0] = l0; s_lp[v1] = l1;
    __syncthreads();                                        // barrier 3

    // Phase E: end-node logsumexp, wave 0 only (no barriers; result on tid 0;
    // overlaps with waves 1..31 running the next step's phases A/B).
    if (tid < 32) {
      float vals[4];
      #pragma unroll
      for (int k = 0; k < 4; ++k) vals[k] = s_lp[s_end[tid + 32 * k]];
      float mE = fmaxf(fmaxf(vals[0], vals[1]), fmaxf(vals[2], vals[3]));
      mE = wave_max32(mE);
      float sE = 0.0f;
      #pragma unroll
      for (int k = 0; k < 4; ++k) sE += pexp(vals[k] - mE);
      sE = wave_sum32(sE);
      if (t + 1 == L && tid == 0) result = mE + __logf(sE) + acc;
    }
  }

  if (tid == 0) outp[b] = -result;
}

// ---------------------------------------------------------------------------
extern "C" void kernel_launch(void* const* d_in, const int* in_sizes, int n_in,
                              void* d_out, int out_size, void* d_ws, size_t ws_size,
                              hipStream_t stream) {
  const float* x    = (const float*)d_in[0];   // [V,B,T]
  const int*   tl   = (const int*)  d_in[1];   // [B]
  const int*   iidx = (const int*)  d_in[2];   // [E]
  const int*   oidx = (const int*)  d_in[3];   // [E]
  const int*   sid  = (const int*)  d_in[4];   // [S]
  const int*   eid  = (const int*)  d_in[5];   // [S]
  float*       out  = (float*)d_out;

  const int B = in_sizes[1];
  const size_t need = (size_t)B * kT * kV * sizeof(float);
  const int use_ws = (ws_size >= need) ? 1 : 0;
  float* xw = (float*)d_ws;

  if (use_ws) {
    dim3 g(kV / 32, kT / 32, B), blk(32, 8);
    transpose_vbt_to_btv<<<g, blk, 0, stream>>>(x, xw, B);
  }
  lattice_forward<<<B, kNT, 0, stream>>>(x, xw, tl, iidx, oidx, sid, eid,
                                         out, B, use_ws);
}